// EfficientMCAttModel_19791209300075
// MI455X (gfx1250) — compile-verified
//
#include <hip/hip_runtime.h>
#include <hip/hip_bf16.h>
#include <math.h>

// Pairwise-distance + mask kernel for EfficientMCAttModel neighbor lists.
//
// Roofline (MI455X): 201 MB of output stores -> ~8.6 us at 23.3 TB/s HBM.
// Inputs (X: 384 KB total, 6 KB per graph) stay resident in WGP$/L2.
// Compute (~0.25 GFLOP) is free => store-bandwidth-bound.
//
// Per graph, the Gram matrix G = Xc * Xc^T (Xc = centered 512x3 coords) is
// computed 16x16-tile-at-a-time with V_WMMA_F32_16X16X4_F32 (K=3 padded to 4),
// then d^2 = |xi|^2 + |xj|^2 - 2*G. Coordinates are centered at the box
// midpoint to minimize cancellation in the Gram identity.
//
// Round-2 changes vs round-1:
//  * row-side shuffles (ds_bpermute), row attributes, and per-row output
//    offsets hoisted out of the column-tile loop (they depend only on v,h)
//  * column-tile count is a template parameter -> full unroll, 4 static WMMAs
//  * 32-bit element offsets for stores (max index 50.3M < 2^31) so the
//    backend can use saddr+voffset stores instead of 64-bit VALU address math

typedef float v2f __attribute__((ext_vector_type(2)));
typedef float v8f __attribute__((ext_vector_type(8)));

#define INTRA_CUTOFF 1.6f   /* 8.0 / 5.0 */
#define INTER_CUTOFF 2.0f   /* 10.0 / 5.0 */
#define CENTER       3.325f /* BOX/2, accuracy shift only (cancels in d^2) */

template <int COLT>
__global__ __launch_bounds__(256)
void pairdist_mask_kernel(const float* __restrict__ X,
                          const int*   __restrict__ segment_id,
                          const int*   __restrict__ is_global,
                          float* __restrict__ out_ctx,
                          float* __restrict__ out_inter,
                          float* __restrict__ out_dist,
                          int maxn, int rowTiles, int colTilesRT)
{
    const int b       = blockIdx.x / rowTiles;    // graph id
    const int rowTile = blockIdx.x % rowTiles;    // 16-row tile within graph
    const int wave    = threadIdx.x >> 5;         // 8 waves / block
    const int lane    = threadIdx.x & 31;
    const int M       = lane & 15;                // row/col slot within tile
    const int h       = lane >> 4;                // lane half (K=2,3 side)

    const int i0   = rowTile * 16;                // local row base
    const int base = b * maxn;                    // global node base of graph

    // ---- A side: this wave's 16 rows (lane L%16 loads full coords of row M)
    const int rowG = base + i0 + M;
    const float px = X[3 * rowG + 0] - CENTER;
    const float py = X[3 * rowG + 1] - CENTER;
    const float pz = X[3 * rowG + 2] - CENTER;
    const float rA_loc = px * px + py * py + pz * pz;
    const int   rattr  = (segment_id[rowG] & 1) | ((is_global[rowG] & 1) << 1);

    // A matrix VGPR layout (16x4, K=3 padded with 0):
    //   lanes 0-15:  VGPR0=K0(x), VGPR1=K1(y);  lanes 16-31: VGPR0=K2(z), VGPR1=K3(0)
    v2f A;
    A.x = h ? pz : px;
    A.y = h ? 0.0f : py;

    // ---- Hoisted per-output-row state (depends only on v and lane half h).
    // C/D layout: VGPR v holds row M=v (lanes 0-15) / M=v+8 (lanes 16-31);
    // the lane that loaded row (v+8h)'s coords/attrs is lane v+8h.
    float    rAv[8];
    int      rsRow[8], rglRow[8], rowvL[8];
    unsigned rowBase[8];                 // 32-bit flat element offset of row
#pragma unroll
    for (int v = 0; v < 8; ++v) {
        const int srcLane = v + 8 * h;
        rAv[v]     = __shfl(rA_loc, srcLane, 32);
        const int rat = __shfl(rattr, srcLane, 32);
        rsRow[v]   = rat & 1;
        rglRow[v]  = (rat >> 1) & 1;
        rowvL[v]   = i0 + srcLane;       // local row index
        rowBase[v] = (unsigned)(base + rowvL[v]) * (unsigned)maxn;
    }

    const int CT = (COLT > 0) ? COLT : colTilesRT;
#pragma unroll
    for (int t = 0; t < CT; ++t) {
        const int j0   = (wave * CT + t) * 16;    // local col base
        const int colL = j0 + M;
        const int colG = base + colL;

        const float qx = X[3 * colG + 0] - CENTER;
        const float qy = X[3 * colG + 1] - CENTER;
        const float qz = X[3 * colG + 2] - CENTER;
        const float rB_loc = qx * qx + qy * qy + qz * qz;
        const int   cs  = segment_id[colG] & 1;
        const int   cgl = is_global[colG] & 1;

        // B matrix (4x16), same half-lane K striping as A.
        v2f Bv;
        Bv.x = h ? qz : qx;
        Bv.y = h ? 0.0f : qy;

        v8f C = {};
        // D = A x B (+0): one 16x16 Gram tile per instruction.
        C = __builtin_amdgcn_wmma_f32_16x16x4_f32(
                /*neg_a=*/false, A, /*neg_b=*/false, Bv,
                /*c_mod=*/(short)0, C, /*reuse_a=*/false, /*reuse_b=*/false);

#pragma unroll
        for (int v = 0; v < 8; ++v) {
            const float g      = C[v];
            const float d2     = rAv[v] + rB_loc - 2.0f * g;
            const bool  validp = (colL != rowvL[v]);
            const bool  finite = validp && (d2 > 0.0f);
            const float dist   = finite ? sqrtf(d2) : INFINITY;

            const int  rs   = rsRow[v];
            const bool rgl  = rglRow[v] != 0;
            const bool cg   = (cgl != 0) && validp;
            const bool notg = !(rgl || cg);
            const bool same = validp && (rs == cs);
            const bool ctxr  = same && (rs == 1) && notg && (dist <= INTRA_CUTOFF);
            const bool inter = validp && (rs != cs) && notg && (dist <= INTER_CUTOFF);
            const bool gn    = same && !notg;
            const bool gg    = validp && rgl && cg;
            const bool ctx   = ctxr || gn || gg;

            const float md = ((ctx || inter) && finite) ? dist : 0.0f;

            const unsigned o = rowBase[v] + (unsigned)colL;  // < 2^31
            out_ctx[o]   = ctx   ? 1.0f : 0.0f;
            out_inter[o] = inter ? 1.0f : 0.0f;
            out_dist[o]  = md;
        }
    }
}

extern "C" void kernel_launch(void* const* d_in, const int* in_sizes, int n_in,
                              void* d_out, int out_size, void* d_ws, size_t ws_size,
                              hipStream_t stream) {
    const float* X          = (const float*)d_in[0];
    const int*   segment_id = (const int*)d_in[2];
    const int*   is_global  = (const int*)d_in[3];

    const int N    = in_sizes[1];               // 32768 nodes
    const int maxn = out_size / (3 * N);        // 512
    const int B    = N / maxn;                  // 64 graphs

    float* out        = (float*)d_out;
    const size_t outN = (size_t)N * (size_t)maxn;
    float* out_ctx    = out;
    float* out_inter  = out + outN;
    float* out_dist   = out + 2 * outN;

    const int rowTiles = maxn / 16;              // 32 row tiles per graph
    const int colTiles = rowTiles / 8;           // col tiles per wave (4)

    dim3 grid(B * rowTiles);                     // 2048 blocks
    dim3 block(256);                             // 8 waves (wave32)
    if (colTiles == 4) {
        pairdist_mask_kernel<4><<<grid, block, 0, stream>>>(
            X, segment_id, is_global, out_ctx, out_inter, out_dist,
            maxn, rowTiles, colTiles);
    } else {
        pairdist_mask_kernel<0><<<grid, block, 0, stream>>>(
            X, segment_id, is_global, out_ctx, out_inter, out_dist,
            maxn, rowTiles, colTiles);
    }
}